// Model_66812511256837
// MI455X (gfx1250) — compile-verified
//
#include <hip/hip_runtime.h>
#include <math.h>

typedef __attribute__((ext_vector_type(2))) float v2f;
typedef __attribute__((ext_vector_type(8))) float v8f;
typedef __attribute__((ext_vector_type(4))) unsigned int v4u;
typedef __attribute__((ext_vector_type(8))) int v8i;
typedef __attribute__((ext_vector_type(4))) int v4i;

#define B_     32
#define L_     1024
#define C_     512
#define MODES_ 64
#define NKK    128   // [o_r | o_i] stacked K for the fused inverse-DFT GEMM
#define KB     32    // K-panel depth staged via TDM in k_final
#define PI_F   3.14159265358979323846f

static __device__ __forceinline__ v8f wmma_f32(v2f a, v2f b, v8f c) {
  // V_WMMA_F32_16X16X4_F32: D = A(16x4) * B(4x16) + C(16x16)
  return __builtin_amdgcn_wmma_f32_16x16x4_f32(false, a, false, b, (short)0, c, false, false);
}

// ---------------------------------------------------------------------------
// TDM: 2D tile load Global->LDS.  D# built per CDNA5 ISA 8.3/8.4:
//  group0: [1:0]=count=1, [63:32]=lds_addr, [120:64]=global_addr, [127:126]=type=2
//  group1: [17:16]=data_size(2=4B), [79:48]=tensor_dim0, [111:80]=tensor_dim1,
//          [127:112]=tile_dim0, [143:128]=tile_dim1, [207:160]=dim0_stride
// All dims/strides in elements; tiles never go OOB here.
// ---------------------------------------------------------------------------
static __device__ __forceinline__ void tdm_load_tile(const float* gsrc, unsigned lds_off,
                                                     unsigned tile_rows, unsigned tile_cols,
                                                     unsigned tensor_w, unsigned tensor_h,
                                                     unsigned row_stride) {
  unsigned long long ga = (unsigned long long)(size_t)gsrc;
  v4u g0;
  g0.x = 1u;                                                   // count=1 (user D#)
  g0.y = lds_off;                                              // lds_addr (bytes)
  g0.z = (unsigned)ga;                                         // global_addr[31:0]
  g0.w = (unsigned)((ga >> 32) & 0x01FFFFFFu) | (2u << 30);    // addr[56:32] | type=2
  v8i g1;
  g1[0] = (int)(2u << 16);                                     // data_size = 4 bytes
  g1[1] = (int)((tensor_w & 0xFFFFu) << 16);                   // tensor_dim0[15:0]
  g1[2] = (int)((tensor_w >> 16) | ((tensor_h & 0xFFFFu) << 16));
  g1[3] = (int)((tensor_h >> 16) | ((tile_cols & 0xFFFFu) << 16)); // tile_dim0
  g1[4] = (int)(tile_rows & 0xFFFFu);                          // tile_dim1 (tile_dim2=0)
  g1[5] = (int)row_stride;                                     // dim0_stride[31:0]
  g1[6] = 0;                                                   // stride[47:32]=0
  g1[7] = 0;
  v4i gz = {0, 0, 0, 0};                                       // 2D: groups 2/3 unused
#if __clang_major__ >= 23
  v8i gz8 = {0, 0, 0, 0, 0, 0, 0, 0};
  __builtin_amdgcn_tensor_load_to_lds(g0, g1, gz, gz, gz8, 0);
#else
  __builtin_amdgcn_tensor_load_to_lds(g0, g1, gz, gz, 0);
#endif
}

// ---------------------------------------------------------------------------
// 1) series_decomp: trend[b,l,c] = movavg25(x along l, replicate pad)
//    x (64MB) is L2-resident (192MB L2), so the 25-tap window stays on-chip.
// ---------------------------------------------------------------------------
__global__ void __launch_bounds__(256) k_decomp(const float* __restrict__ x,
                                                float* __restrict__ trend) {
  int idx = blockIdx.x * blockDim.x + threadIdx.x;            // B*L*C threads
  int c = idx % C_;
  int t = (idx / C_) % L_;
  int b = idx / (C_ * L_);
  const float* xb = x + (size_t)b * L_ * C_ + c;
  float s = 0.f;
  #pragma unroll
  for (int j = -12; j <= 12; ++j) {
    int jj = t + j;
    jj = jj < 0 ? 0 : (jj > L_ - 1 ? L_ - 1 : jj);
    s += xb[(size_t)jj * C_];
  }
  trend[idx] = s * (1.0f / 25.0f);
}

// ---------------------------------------------------------------------------
// 2) DFT basis tables (device-computed, deterministic each call)
// ---------------------------------------------------------------------------
__global__ void __launch_bounds__(256) k_basis_fwd(float* __restrict__ fwdC,
                                                   float* __restrict__ fwdS) {
  int idx = blockIdx.x * blockDim.x + threadIdx.x;            // L*MODES
  int k = idx % MODES_, t = idx / MODES_;
  int r = (k * t) % L_;                                        // exact arg reduction
  float ang = (2.0f * PI_F / L_) * (float)r;
  fwdC[idx] = cosf(ang);                                       // Re(rfft) basis
  fwdS[idx] = -sinf(ang);                                      // Im(rfft) basis
}

__global__ void __launch_bounds__(256) k_basis_inv(float* __restrict__ invB) {
  int idx = blockIdx.x * blockDim.x + threadIdx.x;            // NKK*L
  int t = idx % L_, kk = idx / L_;
  int k = kk & (MODES_ - 1);
  int r = (k * t) % L_;
  float ang = (2.0f * PI_F / L_) * (float)r;
  float v;
  if (kk < MODES_) {                                           // real part of modes
    v = ((k == 0) ? 1.0f : 2.0f) * cosf(ang) * (1.0f / L_);
  } else {                                                     // imag part (DC dropped)
    v = (k == 0) ? 0.0f : (-2.0f * sinf(ang) * (1.0f / L_));
  }
  invB[idx] = v;                                               // invB[kk][t]
}

// ---------------------------------------------------------------------------
// 3) LDS-tiled transposes so every B-fragment load is contiguous along N
// ---------------------------------------------------------------------------
__global__ void __launch_bounds__(256) k_tr_sq(const float* __restrict__ in,
                                               float* __restrict__ outT) {
  // outT[x][y] = in[y][x]; W_trend[p][l] -> WtT[l][p]
  __shared__ float tile[32][33];
  int x0 = blockIdx.x * 32, y0 = blockIdx.y * 32;
  int tx = threadIdx.x & 31, ty = threadIdx.x >> 5;
  #pragma unroll
  for (int r = 0; r < 32; r += 8)
    tile[ty + r][tx] = in[(size_t)(y0 + ty + r) * L_ + x0 + tx];
  __syncthreads();
  #pragma unroll
  for (int r = 0; r < 32; r += 8)
    outT[(size_t)(x0 + ty + r) * L_ + y0 + tx] = tile[tx][ty + r];
}

__global__ void __launch_bounds__(256) k_wT(const float* __restrict__ w,
                                            float* __restrict__ wT) {
  // w[i][o][k] (C,C,MODES) -> wT[k][i][o] (MODES,C,C): kills the 256B-stride reads
  __shared__ float tile[32][33];                               // [o_loc][k_loc]
  int o0 = blockIdx.x * 32;
  int k0 = blockIdx.y * 32;
  int i  = blockIdx.z;
  int tx = threadIdx.x & 31, ty = threadIdx.x >> 5;
  #pragma unroll
  for (int r = 0; r < 32; r += 8)                              // coalesced along k
    tile[ty + r][tx] = w[((size_t)i * C_ + (o0 + ty + r)) * MODES_ + k0 + tx];
  __syncthreads();
  #pragma unroll
  for (int r = 0; r < 32; r += 8)                              // coalesced along o
    wT[((size_t)(k0 + ty + r) * C_ + i) * C_ + o0 + tx] = tile[tx][ty + r];
}

// ---------------------------------------------------------------------------
// 4) Forward DFT GEMM: a_{r,i}[k][b][i] = sum_t (x-trend)[b,t,i] * {cos,-sin}
//    Wave computes 16(M=i) x 64(N=all modes): one A-frag feeds 8 WMMAs.
// ---------------------------------------------------------------------------
__global__ void __launch_bounds__(256) k_fwd(const float* __restrict__ x,
                                             const float* __restrict__ trend,
                                             const float* __restrict__ fwdC,
                                             const float* __restrict__ fwdS,
                                             float* __restrict__ a_r,
                                             float* __restrict__ a_i) {
  int gw   = (blockIdx.x * blockDim.x + threadIdx.x) >> 5;
  int lane = threadIdx.x & 31;
  int half = lane >> 4, lm = lane & 15;
  int mi = gw & 31; int b = gw >> 5;                           // C/16 = 32 M-tiles
  int m0 = mi << 4;
  const float* Xb = x     + (size_t)b * L_ * C_;
  const float* Tb = trend + (size_t)b * L_ * C_;
  v8f accR[4] = {};
  v8f accI[4] = {};
  for (int k0 = 0; k0 < L_; k0 += 4) {
    int ka = k0 + half * 2;
    v2f a;                                                     // res = x - trend
    a.x = Xb[(size_t)ka * C_ + m0 + lm]       - Tb[(size_t)ka * C_ + m0 + lm];
    a.y = Xb[(size_t)(ka + 1) * C_ + m0 + lm] - Tb[(size_t)(ka + 1) * C_ + m0 + lm];
    #pragma unroll
    for (int nt = 0; nt < 4; ++nt) {
      int n = nt * 16 + lm;
      v2f br; br.x = fwdC[ka * MODES_ + n]; br.y = fwdC[(ka + 1) * MODES_ + n];
      v2f bi; bi.x = fwdS[ka * MODES_ + n]; bi.y = fwdS[(ka + 1) * MODES_ + n];
      accR[nt] = wmma_f32(a, br, accR[nt]);
      accI[nt] = wmma_f32(a, bi, accI[nt]);
    }
  }
  #pragma unroll
  for (int nt = 0; nt < 4; ++nt) {
    size_t base = (size_t)(nt * 16 + lm) * (B_ * C_) + (size_t)b * C_ + m0 + half * 8;
    #pragma unroll
    for (int r = 0; r < 8; ++r) { a_r[base + r] = accR[nt][r]; a_i[base + r] = accI[nt][r]; }
  }
}

// ---------------------------------------------------------------------------
// 5) Mode-mix GEMMs: o[b][o][k] = sum_i a[k][b][i] * wT[k][i][o]
//    Wave computes 16(M=b) x 64(N=o): A-frag (strided reads) reused 4x.
// ---------------------------------------------------------------------------
__global__ void __launch_bounds__(256) k_mix(const float* __restrict__ ain,
                                             const float* __restrict__ wT,
                                             float* __restrict__ ocomb,
                                             int kkoff) {
  int gw   = (blockIdx.x * blockDim.x + threadIdx.x) >> 5;
  int lane = threadIdx.x & 31;
  int half = lane >> 4, lm = lane & 15;
  int nb = gw & 7;  gw >>= 3;                                  // C/64 = 8 N-blocks
  int mi = gw & 1;  int k = gw >> 1;                           // B/16 = 2, 64 modes
  int m0 = mi << 4, n0 = nb << 6;
  const float* A  = ain + (size_t)k * B_ * C_;                 // A[m=b][kk=i]
  const float* Bm = wT  + (size_t)k * C_ * C_;                 // B[kk=i][n=o]
  v8f acc[4] = {};
  for (int kk0 = 0; kk0 < C_; kk0 += 4) {
    int ka = kk0 + half * 2;
    v2f a;
    a.x = A[(size_t)(m0 + lm) * C_ + ka];
    a.y = A[(size_t)(m0 + lm) * C_ + ka + 1];
    #pragma unroll
    for (int nt = 0; nt < 4; ++nt) {
      v2f bb;
      bb.x = Bm[(size_t)ka * C_ + n0 + nt * 16 + lm];
      bb.y = Bm[(size_t)(ka + 1) * C_ + n0 + nt * 16 + lm];
      acc[nt] = wmma_f32(a, bb, acc[nt]);
    }
  }
  #pragma unroll
  for (int nt = 0; nt < 4; ++nt) {
    #pragma unroll
    for (int r = 0; r < 8; ++r) {
      int bg = m0 + half * 8 + r;
      ocomb[((size_t)bg * C_ + n0 + nt * 16 + lm) * NKK + k + kkoff] = acc[nt][r];
    }
  }
}

// ---------------------------------------------------------------------------
// 6) Fused final GEMM (dominant, 34 GFLOP):
//      out[b,t,c] = sum_l trend[b,l,c]*W[t,l] + sum_kk o_comb[b,c,kk]*invB[kk,t] + bias[t]
//    Block: 64(M=c) x 64(N=t), 4 waves each owning a 32x32 register tile (2x2 WMMA).
//    B K-panels (Wt^T, 32x64 f32 = 8KB) streamed Global->LDS by the Tensor Data
//    Mover, double-buffered with s_wait_tensorcnt; B-frags then come from LDS,
//    freeing the VMEM pipe for the A stream.
// ---------------------------------------------------------------------------
__global__ void __launch_bounds__(128) k_final(const float* __restrict__ trend,
                                               const float* __restrict__ WtT,
                                               const float* __restrict__ ocomb,
                                               const float* __restrict__ invB,
                                               const float* __restrict__ bias,
                                               float* __restrict__ out) {
  __shared__ float sB[2][KB * 64];                             // 16KB double buffer
  int lane = threadIdx.x & 31;
  int half = lane >> 4, lm = lane & 15;
  int w  = threadIdx.x >> 5;
  int mw = w & 1, nw = w >> 1;                                 // 2x2 wave grid
  int n0blk = blockIdx.x * 64, m0blk = blockIdx.y * 64, b = blockIdx.z;
  int m0 = m0blk + mw * 32, n0 = n0blk + nw * 32;
  int nL = nw * 32 + lm;                                       // LDS column
  const float* Tb = trend + (size_t)b * L_ * C_;
  unsigned lds0 = (unsigned)(size_t)&sB[0][0];
  unsigned lds1 = (unsigned)(size_t)&sB[1][0];
  bool issuer = (__builtin_amdgcn_readfirstlane((int)threadIdx.x) < 32);  // wave 0, scalar branch
  if (issuer) {
    tdm_load_tile(WtT + n0blk,                     lds0, KB, 64, L_, L_, L_);
    tdm_load_tile(WtT + (size_t)KB * L_ + n0blk,   lds1, KB, 64, L_, L_, L_);
  }
  v8f c00 = {}, c01 = {}, c10 = {}, c11 = {};
  const int NCH = L_ / KB;                                     // 32 K-panels
  for (int kt = 0; kt < NCH; ++kt) {
    if (issuer) {
      if (kt < NCH - 1) __builtin_amdgcn_s_wait_tensorcnt(1);  // panel kt arrived
      else              __builtin_amdgcn_s_wait_tensorcnt(0);
    }
    __syncthreads();
    const float* bufp = sB[kt & 1];
    #pragma unroll
    for (int r0 = 0; r0 < KB; r0 += 4) {
      int kl = r0 + half * 2;
      size_t gk = (size_t)kt * KB + kl;
      v2f a0, a1, b0, b1;
      a0.x = Tb[gk * C_ + m0 + lm];        a0.y = Tb[(gk + 1) * C_ + m0 + lm];
      a1.x = Tb[gk * C_ + m0 + 16 + lm];   a1.y = Tb[(gk + 1) * C_ + m0 + 16 + lm];
      b0.x = bufp[kl * 64 + nL];           b0.y = bufp[(kl + 1) * 64 + nL];
      b1.x = bufp[kl * 64 + nL + 16];      b1.y = bufp[(kl + 1) * 64 + nL + 16];
      c00 = wmma_f32(a0, b0, c00);  c01 = wmma_f32(a0, b1, c01);
      c10 = wmma_f32(a1, b0, c10);  c11 = wmma_f32(a1, b1, c11);
    }
    __syncthreads();                                           // buffer free to refill
    if (issuer && (kt + 2 < NCH)) {
      tdm_load_tile(WtT + (size_t)(kt + 2) * KB * L_ + n0blk,
                    (kt & 1) ? lds1 : lds0, KB, 64, L_, L_, L_);
    }
  }
  // inverse-DFT part: K=128, B (invB) is tiny and L2-hot -> direct global frags
  const float* oc = ocomb + (size_t)b * C_ * NKK;
  #pragma unroll 4
  for (int kk0 = 0; kk0 < NKK; kk0 += 4) {
    int ka = kk0 + half * 2;
    v2f a0, a1, b0, b1;
    a0.x = oc[(size_t)(m0 + lm) * NKK + ka];       a0.y = oc[(size_t)(m0 + lm) * NKK + ka + 1];
    a1.x = oc[(size_t)(m0 + 16 + lm) * NKK + ka];  a1.y = oc[(size_t)(m0 + 16 + lm) * NKK + ka + 1];
    b0.x = invB[(size_t)ka * L_ + n0 + lm];        b0.y = invB[(size_t)(ka + 1) * L_ + n0 + lm];
    b1.x = invB[(size_t)ka * L_ + n0 + 16 + lm];   b1.y = invB[(size_t)(ka + 1) * L_ + n0 + 16 + lm];
    c00 = wmma_f32(a0, b0, c00);  c01 = wmma_f32(a0, b1, c01);
    c10 = wmma_f32(a1, b0, c10);  c11 = wmma_f32(a1, b1, c11);
  }
  // bias + write the [B,L,C] transpose straight from the C/D fragment layout
  float bv0 = bias[n0 + lm], bv1 = bias[n0 + 16 + lm];
  size_t base = (size_t)b * L_ * C_;
  size_t ob00 = base + (size_t)(n0 + lm) * C_ + m0 + half * 8;
  size_t ob01 = base + (size_t)(n0 + 16 + lm) * C_ + m0 + half * 8;
  #pragma unroll
  for (int r = 0; r < 8; ++r) {
    out[ob00 + r]      = c00[r] + bv0;
    out[ob01 + r]      = c01[r] + bv1;
    out[ob00 + 16 + r] = c10[r] + bv0;
    out[ob01 + 16 + r] = c11[r] + bv1;
  }
}

// ---------------------------------------------------------------------------
// Host launch
// ---------------------------------------------------------------------------
extern "C" void kernel_launch(void* const* d_in, const int* in_sizes, int n_in,
                              void* d_out, int out_size, void* d_ws, size_t ws_size,
                              hipStream_t stream) {
  (void)in_sizes; (void)n_in; (void)out_size; (void)ws_size;
  const float* x       = (const float*)d_in[0];  // [B,L,C]
  const float* W_trend = (const float*)d_in[1];  // [L,L]  (p,l)
  const float* b_trend = (const float*)d_in[2];  // [L]
  const float* w_real  = (const float*)d_in[3];  // [C,C,MODES]
  const float* w_imag  = (const float*)d_in[4];  // [C,C,MODES]
  float* out = (float*)d_out;

  float* ws = (float*)d_ws;
  size_t off = 0;
  float* trend = ws + off; off += (size_t)B_ * L_ * C_;        // 64 MB
  float* wTr   = ws + off; off += (size_t)MODES_ * C_ * C_;    // 64 MB
  float* wTi   = ws + off; off += (size_t)MODES_ * C_ * C_;    // 64 MB
  float* a_r   = ws + off; off += (size_t)MODES_ * B_ * C_;    // 4 MB
  float* a_i   = ws + off; off += (size_t)MODES_ * B_ * C_;    // 4 MB
  float* ocomb = ws + off; off += (size_t)B_ * C_ * NKK;       // 8 MB
  float* WtT   = ws + off; off += (size_t)L_ * L_;             // 4 MB
  float* fwdC  = ws + off; off += (size_t)L_ * MODES_;
  float* fwdS  = ws + off; off += (size_t)L_ * MODES_;
  float* invB  = ws + off; off += (size_t)NKK * L_;

  // basis tables + decomposition + weight transposes
  k_basis_fwd<<<(L_ * MODES_) / 256, 256, 0, stream>>>(fwdC, fwdS);
  k_basis_inv<<<(NKK * L_) / 256, 256, 0, stream>>>(invB);
  k_decomp<<<(B_ * L_ * C_) / 256, 256, 0, stream>>>(x, trend);
  k_tr_sq<<<dim3(L_ / 32, L_ / 32), 256, 0, stream>>>(W_trend, WtT);
  k_wT<<<dim3(C_ / 32, MODES_ / 32, C_), 256, 0, stream>>>(w_real, wTr);
  k_wT<<<dim3(C_ / 32, MODES_ / 32, C_), 256, 0, stream>>>(w_imag, wTi);

  // forward DFT GEMM: 32 b * 32 M-tiles = 1024 waves, 8 waves/block
  k_fwd<<<(B_ * (C_ / 16)) / 8, 256, 0, stream>>>(x, trend, fwdC, fwdS, a_r, a_i);

  // mode-mix GEMMs: 64 modes * 2 M-tiles * 8 N-blocks = 1024 waves per half
  k_mix<<<(MODES_ * (B_ / 16) * (C_ / 64)) / 8, 256, 0, stream>>>(a_r, wTr, ocomb, 0);
  k_mix<<<(MODES_ * (B_ / 16) * (C_ / 64)) / 8, 256, 0, stream>>>(a_i, wTi, ocomb, MODES_);

  // fused trend + inverse-DFT GEMM with TDM-staged B panels
  k_final<<<dim3(L_ / 64, C_ / 64, B_), 128, 0, stream>>>(
      trend, WtT, ocomb, invB, b_trend, out);
}